// QLP_14731737825362
// MI455X (gfx1250) — compile-verified
//
#include <hip/hip_runtime.h>

typedef __attribute__((ext_vector_type(2))) float v2f;
typedef __attribute__((ext_vector_type(8))) float v8f;

#define N_WIRES 8
#define DIM 256
#define STATES_PER_BLOCK 16
#define THREADS (STATES_PER_BLOCK * 32)
#define LDS_STRIDE 260   // 256 + 4 floats pad -> conflict-free column reads

// Apply a block of 8 shared-parameter RY gates to the in-register state.
// Wire i acts on bit position p = 7-i of the basis index (idx = r*32+lane).
__device__ __forceinline__ void apply_ry_block(float re[8], float im[8],
                                               const float* __restrict__ theta,
                                               int lane) {
#pragma unroll
  for (int i = 0; i < N_WIRES; ++i) {
    float s, c;
    __sincosf(theta[i] * 0.5f, &s, &c);
    const int p = 7 - i;                    // compile-time (loop unrolled)
    if (p < 5) {
      // bit lives in the lane id: cross-lane butterfly via shfl_xor
      const int mask = 1 << p;
      const float sgn = ((lane >> p) & 1) ? s : -s;
#pragma unroll
      for (int r = 0; r < 8; ++r) {
        float pre = __shfl_xor(re[r], mask, 32);
        float pim = __shfl_xor(im[r], mask, 32);
        re[r] = fmaf(c, re[r], sgn * pre);
        im[r] = fmaf(c, im[r], sgn * pim);
      }
    } else {
      // bit lives in the register index: in-lane butterfly
      const int m = 1 << (p - 5);
#pragma unroll
      for (int r = 0; r < 8; ++r) {
        if (!(r & m)) {
          const int r2 = r | m;
          float ar = re[r], ai = im[r], br = re[r2], bi = im[r2];
          re[r]  = c * ar - s * br;  im[r]  = c * ai - s * bi;
          re[r2] = s * ar + c * br;  im[r2] = s * ai + c * bi;
        }
      }
    }
  }
}

__global__ __launch_bounds__(THREADS)
void vqc_kernel(const float* __restrict__ x,     // [16384, 8]
                const float* __restrict__ ry1,   // [8]
                const float* __restrict__ crz1,  // [8]
                const float* __restrict__ ry2,   // [8]
                const float* __restrict__ crz2,  // [8]
                float* __restrict__ out)         // [16384, 8]
{
  __shared__ float probs_lds[STATES_PER_BLOCK * LDS_STRIDE];

  const int tid   = threadIdx.x;
  const int wave  = tid >> 5;
  const int lane  = tid & 31;
  const int state = blockIdx.x * STATES_PER_BLOCK + wave;

  // ---------------- encoder: product state amplitudes -------------------
  float cs[N_WIRES], sn[N_WIRES];
#pragma unroll
  for (int i = 0; i < N_WIRES; ++i) {
    float xi = x[state * N_WIRES + i] * 0.5f;
    __sincosf(xi, &sn[i], &cs[i]);
  }
  float lanefac = 1.0f;
#pragma unroll
  for (int p = 0; p < 5; ++p) {               // lane bits -> wires 7..3
    const int i = 7 - p;
    lanefac *= ((lane >> p) & 1) ? sn[i] : cs[i];
  }
  float re[8], im[8];
#pragma unroll
  for (int r = 0; r < 8; ++r) {
    float f = lanefac;
#pragma unroll
    for (int q = 0; q < 3; ++q) {             // register bits -> wires 2..0
      const int i = 7 - (5 + q);
      f *= ((r >> q) & 1) ? sn[i] : cs[i];
    }
    re[r] = f;
    im[r] = 0.0f;
  }

  // ---------------- CRZ block 1: ctrl=i, tgt=(i+1)%8 --------------------
  {
    float th[N_WIRES];
#pragma unroll
    for (int i = 0; i < N_WIRES; ++i) th[i] = crz1[i];
#pragma unroll
    for (int r = 0; r < 8; ++r) {
      const int idx = (r << 5) | lane;
      float expo = 0.0f;
#pragma unroll
      for (int i = 0; i < N_WIRES; ++i) {
        const float bc = (float)((idx >> (7 - i)) & 1);
        const float bt = (float)((idx >> (7 - ((i + 1) & 7))) & 1);
        expo += bc * (bt - 0.5f) * th[i];
      }
      float sp, cp;
      __sincosf(expo, &sp, &cp);
      im[r] = re[r] * sp;      // state was purely real after the encoder
      re[r] = re[r] * cp;
    }
  }

  // ---------------- RY block 1 ------------------------------------------
  {
    float th[N_WIRES];
#pragma unroll
    for (int i = 0; i < N_WIRES; ++i) th[i] = ry1[i];
    apply_ry_block(re, im, th, lane);
  }

  // ---------------- CRZ block 2: ctrl=i, tgt=(i-1)%8 --------------------
  {
    float th[N_WIRES];
#pragma unroll
    for (int i = 0; i < N_WIRES; ++i) th[i] = crz2[i];
#pragma unroll
    for (int r = 0; r < 8; ++r) {
      const int idx = (r << 5) | lane;
      float expo = 0.0f;
#pragma unroll
      for (int i = 0; i < N_WIRES; ++i) {
        const float bc = (float)((idx >> (7 - i)) & 1);
        const float bt = (float)((idx >> (7 - ((i + 7) & 7))) & 1);
        expo += bc * (bt - 0.5f) * th[i];
      }
      float sp, cp;
      __sincosf(expo, &sp, &cp);
      const float nr = re[r] * cp - im[r] * sp;
      im[r] = re[r] * sp + im[r] * cp;
      re[r] = nr;
    }
  }

  // ---------------- RY block 2 ------------------------------------------
  {
    float th[N_WIRES];
#pragma unroll
    for (int i = 0; i < N_WIRES; ++i) th[i] = ry2[i];
    apply_ry_block(re, im, th, lane);
  }

  // ---------------- probabilities -> LDS --------------------------------
#pragma unroll
  for (int r = 0; r < 8; ++r) {
    const int idx = (r << 5) | lane;
    probs_lds[wave * LDS_STRIDE + idx] = re[r] * re[r] + im[r] * im[r];
  }
  __syncthreads();

  // ---------------- readout GEMM via WMMA (wave 0 only) -----------------
  // C(16 states x 16 cols) = probs(16 x 256) @ signs(256 x 16),
  // signs[k][n] = 1 - 2*bit(k, 7-n) for n<8, 0 for padding cols 8..15.
  if (wave == 0) {
    const int M    = lane & 15;          // A row (state within block)
    const int N    = lane & 15;          // B column (output wire, padded)
    const int koff = (lane >> 4) * 2;    // lower lanes K=0,1; upper K=2,3

    // Branchless B-operand generation: sign bits OR'd onto 1.0f's pattern,
    // zeroed entirely for the padding columns. No exec-mask traffic.
    const unsigned validmask = (N < 8) ? 0xFFFFFFFFu : 0u;
    const int      shift     = 7 - (N & 7);

    v8f acc0 = {};
    v8f acc1 = {};                       // dual accumulators: break WMMA RAW chain
#pragma unroll 8
    for (int kt = 0; kt < DIM / 4; ++kt) {
      const int kbase = kt * 4 + koff;
      v2f a;
      a.x = probs_lds[M * LDS_STRIDE + kbase];
      a.y = probs_lds[M * LDS_STRIDE + kbase + 1];
      const unsigned bx = ((unsigned)kbase >> shift) & 1u;
      const unsigned by = ((unsigned)(kbase + 1) >> shift) & 1u;
      v2f b;
      b.x = __uint_as_float((0x3F800000u | (bx << 31)) & validmask);
      b.y = __uint_as_float((0x3F800000u | (by << 31)) & validmask);
      if (kt & 1) {
        acc1 = __builtin_amdgcn_wmma_f32_16x16x4_f32(
            false, a, false, b, (short)0, acc1, false, false);
      } else {
        acc0 = __builtin_amdgcn_wmma_f32_16x16x4_f32(
            false, a, false, b, (short)0, acc0, false, false);
      }
    }
    v8f acc = acc0 + acc1;

    // C/D layout: VGPR v, lanes 0-15 -> M=v, N=lane; lanes 16-31 -> M=v+8.
    const int mbase = (lane >> 4) * 8;
    const int n     = lane & 15;
    if (n < 8) {
      const int s0 = blockIdx.x * STATES_PER_BLOCK;
#pragma unroll
      for (int v = 0; v < 8; ++v) {
        out[(s0 + mbase + v) * N_WIRES + n] = acc[v];
      }
    }
  }
}

extern "C" void kernel_launch(void* const* d_in, const int* in_sizes, int n_in,
                              void* d_out, int out_size, void* d_ws, size_t ws_size,
                              hipStream_t stream) {
  const float* x    = (const float*)d_in[0];   // input_features (4,4096,8)
  const float* ry1  = (const float*)d_in[1];   // params_ry1 (1,8)
  const float* crz1 = (const float*)d_in[2];   // params_crz1 (1,8)
  const float* ry2  = (const float*)d_in[3];   // params_ry2 (1,8)
  const float* crz2 = (const float*)d_in[4];   // params_crz2 (1,8)
  float* out = (float*)d_out;

  const int n_states = in_sizes[0] / N_WIRES;          // 16384
  const int blocks   = n_states / STATES_PER_BLOCK;    // 1024
  vqc_kernel<<<blocks, THREADS, 0, stream>>>(x, ry1, crz1, ry2, crz2, out);
}